// LAT_20306605375901
// MI455X (gfx1250) — compile-verified
//
#include <hip/hip_runtime.h>

// CDNA5 / gfx1250 3-D neighborhood attention (NATTEN-style).
//  - QK logits: V_WMMA_F32_16X16X4_F32 band-GEMMs (14-query tiles -> the
//    3-wide diagonal band fits in 16 key columns, no edge cases).
//  - K/Q staging: contiguous 6KB slabs DMA'd with global_load_async_to_lds_b128,
//    double-buffered, synced via s_wait_asynccnt + workgroup barriers.
//  - Band extraction via a per-wave LDS bounce fenced with s_wait_dscnt.

typedef __attribute__((ext_vector_type(2))) float v2f;
typedef __attribute__((ext_vector_type(4))) float v4f;
typedef __attribute__((ext_vector_type(8))) float v8f;

#define DIM   40
#define CH    96
#define NHEAD 6
#define HD    16
#define TQ    14          // queries per tile (band fits in 16 key columns)
#define NTILE 3           // 3*14 = 42 >= 40
#define NVOX  (DIM*DIM*DIM)
#define SLABD (16*CH)     // 1536 dwords = 6KB: 16 rows x 96 channels

__device__ __forceinline__ int clampi(int x) {
  return x < 0 ? 0 : (x > DIM - 1 ? DIM - 1 : x);
}

__device__ __forceinline__ v8f wmma_f32_k4(v2f a, v2f b, v8f c) {
  // D = A(16x4) * B(4x16) + C, f32, wave32
  return __builtin_amdgcn_wmma_f32_16x16x4_f32(
      /*neg_a=*/false, a, /*neg_b=*/false, b,
      /*c_mod=*/(short)0, c, /*reuse_a=*/false, /*reuse_b=*/false);
}

// Async-DMA one contiguous 16-row x 96-ch f32 slab (6KB) into LDS.
// Rows are clamped so every lane always has a legal address; validity is
// re-applied (as zeros) on the read side.  Always issues exactly 2 async
// instructions per wave so ASYNCcnt bookkeeping is uniform.
__device__ __forceinline__ void slab_load_async(const float* __restrict__ base,
                                                int plane,   // clamped hh*40+ww
                                                int kt0,     // first row t (may be -1)
                                                float* slab, int tid) {
  #pragma unroll
  for (int r = 0; r < 2; ++r) {
    const int d   = r * 768 + tid * 4;          // dword index in slab
    const int row = d / CH;
    const int ch  = d - row * CH;
    const int kt  = clampi(kt0 + row);
    const float* gp = base + (size_t)(plane * DIM + kt) * CH + ch;
    const unsigned int lds_lo = (unsigned int)(uintptr_t)(slab + d);
    asm volatile("global_load_async_to_lds_b128 %0, %1, off"
                 :: "v"(lds_lo), "v"((unsigned long long)(uintptr_t)gp)
                 : "memory");
  }
}

__global__ __launch_bounds__(NHEAD * 32)
void natt3d_wmma(const float* __restrict__ q,
                 const float* __restrict__ kmat,
                 const float* __restrict__ rpb,
                 float* __restrict__ out)
{
  __shared__ float qsl[SLABD];          // 6KB  query slab
  __shared__ float ksl[2][SLABD];       // 12KB key slab ping-pong
  __shared__ float dbuf[NHEAD * 256];   // 6KB  per-wave D-matrix bounce

  const int tid  = threadIdx.x;
  const int wave = tid >> 5;            // == head index
  const int lane = tid & 31;
  const int m    = lane & 15;           // query row (A) / key column (B)
  const int hi   = lane >> 4;           // half-wave selects K sub-dims

  int b = blockIdx.x;
  const int tt = b % NTILE;  b /= NTILE;
  const int w  = b % DIM;
  const int h  = b / DIM;
  const int t0 = tt * TQ;               // 0, 14, 28

  // ---- stage Q slab and K slab 0 ----------------------------------------
  slab_load_async(q, h * DIM + w, t0, qsl, tid);
  slab_load_async(kmat, clampi(h - 1) * DIM + clampi(w - 1), t0 - 1, ksl[0], tid);

  asm volatile("s_wait_asynccnt 0x2" ::: "memory");  // Q slab landed
  __syncthreads();

  // ---- build A operands: 16 queries x 16 dims, pre-scaled ----------------
  // f32 16x16x4 A layout: lanes 0-15 hold (K=4kk+0,4kk+1) of row m,
  // lanes 16-31 hold (K=4kk+2,4kk+3) of row m  -> channel = 4kk + 2*hi.
  const int  tq = t0 + m;
  const bool qv = (m < TQ) && (tq < DIM);
  const v2f  z2 = {0.f, 0.f};
  const float sc = 0.25f;                          // hd^-0.5, hd = 16
  const float* qp = &qsl[m * CH + wave * HD + hi * 2];
  v2f A0 = *(const v2f*)(qp +  0) * sc;
  v2f A1 = *(const v2f*)(qp +  4) * sc;
  v2f A2 = *(const v2f*)(qp +  8) * sc;
  v2f A3 = *(const v2f*)(qp + 12) * sc;
  A0 = qv ? A0 : z2;  A1 = qv ? A1 : z2;
  A2 = qv ? A2 : z2;  A3 = qv ? A3 : z2;

  float* wl = &dbuf[wave * 256];
  const int mq  = (m < TQ) ? m : (TQ - 1);         // clamp for safe LDS reads
  const int off = ((mq >> 3) << 7) + (mq & 7);     // half*16lanes*8 + vgpr idx

  float lg[27];

  // ---- 9 (di,dj) planes: one 16x16x16 band-GEMM each ---------------------
  #pragma unroll
  for (int dij = 0; dij < 9; ++dij) {
    const int p = dij & 1;

    // prefetch next K slab into the other buffer, then wait for current
    if (dij < 8) {
      const int nd = dij + 1;
      const int plane = clampi(h + nd / 3 - 1) * DIM + clampi(w + nd % 3 - 1);
      slab_load_async(kmat, plane, t0 - 1, ksl[p ^ 1], tid);
      asm volatile("s_wait_asynccnt 0x2" ::: "memory");  // current slab done
    } else {
      asm volatile("s_wait_asynccnt 0x0" ::: "memory");
    }
    __syncthreads();                                     // visible to all waves

    // B operands from LDS; zero out-of-range keys (reference zero-padding)
    const int di = dij / 3 - 1, dj = dij % 3 - 1;
    const bool plane_ok = ((unsigned)(h + di) < (unsigned)DIM) &&
                          ((unsigned)(w + dj) < (unsigned)DIM);
    const bool kv = plane_ok && ((unsigned)(t0 - 1 + m) < (unsigned)DIM);
    const float* kp = &ksl[p][m * CH + wave * HD + hi * 2];
    v2f B0 = *(const v2f*)(kp +  0);
    v2f B1 = *(const v2f*)(kp +  4);
    v2f B2 = *(const v2f*)(kp +  8);
    v2f B3 = *(const v2f*)(kp + 12);
    B0 = kv ? B0 : z2;  B1 = kv ? B1 : z2;
    B2 = kv ? B2 : z2;  B3 = kv ? B3 : z2;

    __syncthreads();   // all waves done reading ksl[p]; safe to overwrite next iter

    v8f acc = {0.f, 0.f, 0.f, 0.f, 0.f, 0.f, 0.f, 0.f};
    acc = wmma_f32_k4(A0, B0, acc);
    acc = wmma_f32_k4(A1, B1, acc);
    acc = wmma_f32_k4(A2, B2, acc);
    acc = wmma_f32_k4(A3, B3, acc);

    // D layout: lane l, VGPR r <-> D[r + 8*(l>>4), l&15]; bounce via LDS.
    v4f lo = __builtin_shufflevector(acc, acc, 0, 1, 2, 3);
    v4f hv = __builtin_shufflevector(acc, acc, 4, 5, 6, 7);
    *(v4f*)(wl + lane * 8)     = lo;
    *(v4f*)(wl + lane * 8 + 4) = hv;

    asm volatile("s_wait_dscnt 0" ::: "memory");   // wave-local fence

    // band extract: query m needs columns m, m+1, m+2 (dl = -1, 0, +1)
    #pragma unroll
    for (int s = 0; s < 3; ++s) {
      const int c = mq + s;                        // <= 15 by construction
      lg[dij * 3 + s] = wl[c * 8 + off];
    }
    asm volatile("s_wait_dscnt 0" ::: "memory");   // WAR fence before next store
  }

  // ---- +rpb, softmax(27), expected offset --------------------------------
  float mx = -3.4e38f;
  #pragma unroll
  for (int n = 0; n < 27; ++n) {
    lg[n] += rpb[wave * 27 + n];                   // wave-uniform -> s_load
    mx = fmaxf(mx, lg[n]);
  }
  float sum = 0.f;
  #pragma unroll
  for (int n = 0; n < 27; ++n) {
    const float e = __expf(lg[n] - mx);            // v_exp_f32
    lg[n] = e;
    sum += e;
  }
  const float inv = 1.0f / sum;

  float x0 = 0.f, x1 = 0.f, x2 = 0.f;
  #pragma unroll
  for (int n = 0; n < 27; ++n) {
    const float a = lg[n] * inv;
    x0 += a * (float)(n / 9 - 1);                  // di offset
    x1 += a * (float)((n / 3) % 3 - 1);            // dj offset
    x2 += a * (float)(n % 3 - 1);                  // dl offset
  }

  // out: [1, NH*3, 40, 40, 40], channel = head*3 + coord
  if (hi == 0 && m < TQ && tq < DIM) {
    const int sp = (h * DIM + w) * DIM + tq;
    out[(wave * 3 + 0) * NVOX + sp] = x0;
    out[(wave * 3 + 1) * NVOX + sp] = x1;
    out[(wave * 3 + 2) * NVOX + sp] = x2;
  }
}

extern "C" void kernel_launch(void* const* d_in, const int* in_sizes, int n_in,
                              void* d_out, int out_size, void* d_ws, size_t ws_size,
                              hipStream_t stream) {
  (void)in_sizes; (void)n_in; (void)out_size; (void)d_ws; (void)ws_size;
  const float* q   = (const float*)d_in[0];
  const float* k   = (const float*)d_in[1];
  const float* rpb = (const float*)d_in[2];
  float* out = (float*)d_out;

  dim3 grid(DIM * DIM * NTILE);   // 4800 workgroups
  dim3 block(NHEAD * 32);         // 6 wave32s, one head each
  natt3d_wmma<<<grid, block, 0, stream>>>(q, k, rpb, out);
}